// Mamba_44779329028514
// MI455X (gfx1250) — compile-verified
//
#include <hip/hip_runtime.h>
#include <hip/hip_bf16.h>
#include <math.h>

#define D_MODEL 1024
#define D_STATE 16
#define D_CONV  4
#define D_INNER 2048
#define DT_RANK 64
#define BATCH   2
#define SEQLEN  4096
#define NROWS   (BATCH * SEQLEN)   // 8192

typedef __attribute__((ext_vector_type(2))) float v2f;
typedef __attribute__((ext_vector_type(4))) float v4f;
typedef __attribute__((ext_vector_type(8))) float v8f;
typedef __attribute__((ext_vector_type(4))) int   v4i;

#if defined(__gfx1250__) && __has_builtin(__builtin_amdgcn_global_load_async_to_lds_b128) && __has_builtin(__builtin_amdgcn_s_wait_asynccnt)
#define HAVE_ASYNC_LDS 1
typedef __attribute__((address_space(1))) v4i* gas_v4i_ptr;
typedef __attribute__((address_space(3))) v4i* lds_v4i_ptr;
#else
#define HAVE_ASYNC_LDS 0
#endif

__device__ __forceinline__ float sigmoidf_(float x) { return 1.0f / (1.0f + __expf(-x)); }
__device__ __forceinline__ float softplusf_(float x) { return (x > 20.0f) ? x : log1pf(__expf(x)); }

// ---------------------------------------------------------------------------
// f32 WMMA GEMM:  C[M,N] = act( A[M,K] * W[N,K]^T (+ bias[n]) )
// One wave -> (16*MT) x (16*NT) output tile via V_WMMA_F32_16X16X4_F32.
// K is consumed 8 at a time: each lane loads a b128 (4 consecutive K values at
// k + 4*(lane>>4)); the .xy pair feeds WMMA #1 (hardware K-slots {0,1}/{2,3}
// map to actual K {k,k+1}/{k+4,k+5}), the .zw pair feeds WMMA #2
// (actual K {k+2,k+3}/{k+6,k+7}). A and B use identical per-lane K mappings,
// so the contraction covers each K exactly once.
// EPI: 0 = none, 1 = softplus(x + bias[col])
// ---------------------------------------------------------------------------
template <int MT, int NT, int EPI>
__global__ __launch_bounds__(128) void gemm_wmma_f32(
    const float* __restrict__ A, int lda,
    const float* __restrict__ W, int ldw,
    float* __restrict__ C, int ldc,
    int K, const float* __restrict__ bias)
{
    const int lane  = threadIdx.x & 31;
    const int wave  = threadIdx.x >> 5;
    const int mgrp  = blockIdx.y * 4 + wave;        // group of MT row-tiles
    const int nbase = blockIdx.x * (16 * NT);
    const int r     = lane & 15;
    const int half  = lane >> 4;
    const int koff  = half << 2;                    // 0 or 4

    const float* Ap[MT];
#pragma unroll
    for (int i = 0; i < MT; ++i)
        Ap[i] = A + (size_t)(mgrp * (16 * MT) + i * 16 + r) * lda + koff;
    const float* Wp[NT];
#pragma unroll
    for (int j = 0; j < NT; ++j)
        Wp[j] = W + (size_t)(nbase + j * 16 + r) * ldw + koff;

    v8f acc[MT][NT] = {};

    for (int k = 0; k < K; k += 8) {
        v4f bq[NT];
#pragma unroll
        for (int j = 0; j < NT; ++j)
            bq[j] = *(const v4f*)(Wp[j] + k);
        v4f aq[MT];
#pragma unroll
        for (int i = 0; i < MT; ++i) {
            aq[i] = *(const v4f*)(Ap[i] + k);
            if (k + 128 < K) __builtin_prefetch(Ap[i] + k + 128, 0, 1);
        }
#pragma unroll
        for (int i = 0; i < MT; ++i) {
            const v2f alo = __builtin_shufflevector(aq[i], aq[i], 0, 1);
            const v2f ahi = __builtin_shufflevector(aq[i], aq[i], 2, 3);
#pragma unroll
            for (int j = 0; j < NT; ++j) {
                const v2f blo = __builtin_shufflevector(bq[j], bq[j], 0, 1);
                const v2f bhi = __builtin_shufflevector(bq[j], bq[j], 2, 3);
                acc[i][j] = __builtin_amdgcn_wmma_f32_16x16x4_f32(
                    false, alo, false, blo, (short)0, acc[i][j], false, false);
                acc[i][j] = __builtin_amdgcn_wmma_f32_16x16x4_f32(
                    false, ahi, false, bhi, (short)0, acc[i][j], false, false);
            }
        }
    }

    // C/D layout: VGPR v -> M = v + 8*(lane>>4), N = lane&15
#pragma unroll
    for (int i = 0; i < MT; ++i) {
#pragma unroll
        for (int j = 0; j < NT; ++j) {
            const int col = nbase + j * 16 + r;
#pragma unroll
            for (int v = 0; v < 8; ++v) {
                const int row = mgrp * (16 * MT) + i * 16 + (half << 3) + v;
                float x = acc[i][j][v];
                if (EPI == 1) x = softplusf_(x + bias[col]);
                C[(size_t)row * ldc + col] = x;
            }
        }
    }
}

// ---------------------------------------------------------------------------
// Depthwise causal conv1d (width 4) + SiLU.
// Reads x-half of XZ (stride 4096), writes XC (stride 2048).
// ---------------------------------------------------------------------------
__global__ __launch_bounds__(256) void conv_silu_kernel(
    const float* __restrict__ XZ,
    const float* __restrict__ cw,
    const float* __restrict__ cb,
    float* __restrict__ XC)
{
    size_t idx = (size_t)blockIdx.x * blockDim.x + threadIdx.x;  // over NROWS*D_INNER
    const int d = (int)(idx & (D_INNER - 1));
    const size_t bl = idx >> 11;                 // b*L + l   (D_INNER == 2^11)
    const int l = (int)(bl & (SEQLEN - 1));      // SEQLEN is a power of two

    const float* w = cw + d * D_CONV;
    float acc = cb[d];
#pragma unroll
    for (int k = 0; k < D_CONV; ++k) {
        const int ll = l + k - (D_CONV - 1);
        if (ll >= 0) {
            const size_t row = (size_t)((long)bl + (k - (D_CONV - 1)));
            acc += w[k] * XZ[row * 4096 + d];
        }
    }
    XC[bl * D_INNER + d] = acc * sigmoidf_(acc);
}

// ---------------------------------------------------------------------------
// Selective scan. One wave owns 2 channels; each lane owns one (channel, state)
// pair: sub = lane>>4 selects the channel, n = lane&15 the state.
// h_t = exp(delta*A)*h + (delta*u)*B_t[n]; y = sum_n h*C_t[n] (shfl_xor tree).
// The per-step 32-float (B_t|C_t) block is double-buffered into LDS with
// GLOBAL_LOAD_ASYNC_TO_LDS_B128 (lanes 0..7 issue, s_wait_asynccnt 1 pipelines
// the t+1 fetch under step t's math). Lane n==0 applies D-skip + SiLU(z) gate
// and writes y over the dead x-half of XZ (col d); z lives at col 2048+d.
// ---------------------------------------------------------------------------
__global__ __launch_bounds__(256) void scan_kernel(
    const float* __restrict__ DEL,
    const float* __restrict__ XC,
    const float* __restrict__ DBL,
    const float* __restrict__ A_log,
    const float* __restrict__ Dvec,
    float* __restrict__ XZ)
{
    const int lane = threadIdx.x & 31;
    const int wv   = threadIdx.x >> 5;
    const int b    = blockIdx.y;
    const int dpair = blockIdx.x * 8 + wv;      // 2 channels per wave
    const int sub  = lane >> 4;
    const int n    = lane & 15;
    const int d    = dpair * 2 + sub;

    const float a  = -__expf(A_log[d * D_STATE + n]);
    const float Dd = Dvec[d];
    float h = 0.0f;

    const size_t baseDbl = (size_t)b * SEQLEN * 96;
    const size_t baseD   = (size_t)b * SEQLEN * (size_t)D_INNER;
    const size_t baseXZ  = (size_t)b * SEQLEN * 4096;

#if HAVE_ASYNC_LDS
    __shared__ float smBC[8][2][32];
    // lanes 0..7: each fetches 16B of the 128B (B_t|C_t) block for step t.
    auto issue = [&](int t, int buf) {
        if (lane < 8) {
            const float* g = DBL + baseDbl + (size_t)t * 96 + DT_RANK + lane * 4;
            __builtin_amdgcn_global_load_async_to_lds_b128(
                (gas_v4i_ptr)g,
                (lds_v4i_ptr)&smBC[wv][buf][lane * 4],
                0, 0);
        }
    };
    issue(0, 0);
#endif

    for (int t = 0; t < SEQLEN; ++t) {
        const size_t rowD = baseD + (size_t)t * D_INNER;
        const float delta = DEL[rowD + d];
        const float u     = XC[rowD + d];

#if HAVE_ASYNC_LDS
        const int tn = (t + 1 < SEQLEN) ? t + 1 : t;   // tail refetch keeps wait imm constant
        issue(tn, (t + 1) & 1);
        __builtin_amdgcn_s_wait_asynccnt(1);
        asm volatile("" ::: "memory");
        const float Bn = smBC[wv][t & 1][n];
        const float Cn = smBC[wv][t & 1][16 + n];
#else
        const float* pbl = DBL + baseDbl + (size_t)t * 96;
        const float Bn = pbl[DT_RANK + n];
        const float Cn = pbl[DT_RANK + D_STATE + n];
#endif

        h = __expf(delta * a) * h + (delta * u) * Bn;

        float p = h * Cn;
        p += __shfl_xor(p, 8, 16);
        p += __shfl_xor(p, 4, 16);
        p += __shfl_xor(p, 2, 16);
        p += __shfl_xor(p, 1, 16);

        if (n == 0) {
            const float y  = p + u * Dd;
            const size_t rowXZ = baseXZ + (size_t)t * 4096;
            const float zv = XZ[rowXZ + D_INNER + d];
            XZ[rowXZ + d]  = y * (zv * sigmoidf_(zv));
        }
    }
}

// ---------------------------------------------------------------------------
extern "C" void kernel_launch(void* const* d_in, const int* in_sizes, int n_in,
                              void* d_out, int out_size, void* d_ws, size_t ws_size,
                              hipStream_t stream)
{
    (void)in_sizes; (void)n_in; (void)out_size; (void)ws_size;

    const float* hs     = (const float*)d_in[0];  // (B, L, 1024)
    const float* W_in   = (const float*)d_in[1];  // (4096, 1024)
    const float* conv_w = (const float*)d_in[2];  // (2048, 4)
    const float* conv_b = (const float*)d_in[3];  // (2048,)
    const float* W_x    = (const float*)d_in[4];  // (96, 2048)
    const float* W_dt   = (const float*)d_in[5];  // (2048, 64)
    const float* dt_b   = (const float*)d_in[6];  // (2048,)
    const float* A_log  = (const float*)d_in[7];  // (2048, 16)
    const float* Dvec   = (const float*)d_in[8];  // (2048,)
    const float* W_out  = (const float*)d_in[9];  // (1024, 2048)
    float* out = (float*)d_out;                   // (B, L, 1024)

    float* ws  = (float*)d_ws;
    float* XZ  = ws;                                   // NROWS * 4096  (x | z)
    float* XC  = XZ  + (size_t)NROWS * 4096;           // NROWS * 2048
    float* DBL = XC  + (size_t)NROWS * (size_t)D_INNER;// NROWS * 96
    float* DEL = DBL + (size_t)NROWS * 96;             // NROWS * 2048

    // 1) xz = hs @ W_in^T   (M=8192, K=1024, N=4096); wave tile 32x64
    gemm_wmma_f32<2, 4, 0><<<dim3(4096 / 64, NROWS / 128), 128, 0, stream>>>(
        hs, D_MODEL, W_in, D_MODEL, XZ, 4096, D_MODEL, nullptr);

    // 2) depthwise causal conv1d + SiLU
    conv_silu_kernel<<<(NROWS * D_INNER) / 256, 256, 0, stream>>>(
        XZ, conv_w, conv_b, XC);

    // 3) x_dbl = xc @ W_x^T  (M=8192, K=2048, N=96); wave tile 32x96
    gemm_wmma_f32<2, 6, 0><<<dim3(1, NROWS / 128), 128, 0, stream>>>(
        XC, D_INNER, W_x, D_INNER, DBL, 96, D_INNER, nullptr);

    // 4) delta = softplus(dt @ W_dt^T + dt_bias)  (M=8192, K=64, N=2048)
    gemm_wmma_f32<2, 4, 1><<<dim3(D_INNER / 64, NROWS / 128), 128, 0, stream>>>(
        DBL, 96, W_dt, DT_RANK, DEL, D_INNER, DT_RANK, dt_b);

    // 5) selective scan + D-skip + SiLU(z) gate; writes y into XZ[:, 0:2048]
    scan_kernel<<<dim3(D_INNER / 16, BATCH), 256, 0, stream>>>(
        DEL, XC, DBL, A_log, Dvec, XZ);

    // 6) out = y @ W_out^T  (M=8192, K=2048, N=1024)
    gemm_wmma_f32<2, 4, 0><<<dim3(1024 / 64, NROWS / 128), 128, 0, stream>>>(
        XZ, 4096, W_out, D_INNER, out, D_MODEL, D_INNER, nullptr);
}